// TemporalMambaBlock_52656299049431
// MI455X (gfx1250) — compile-verified
//
#include <hip/hip_runtime.h>
#include <hip/hip_bf16.h>

// ---------------------------------------------------------------------------
// TemporalMambaBlock on gfx1250 (MI455X).
// Einsums on V_WMMA_F32_16X16X4_F32 (exact fp32 matrix pipe).
// Weight tiles for the two large GEMMs staged LDS-side via the Tensor Data
// Mover (TENSOR_LOAD_TO_LDS, per-wave double buffer, TENSORcnt-tracked).
// B=2, L=2048, D=768, d_inner=1536, N=16, R=48.  M = B*L = 4096 rows.
// ---------------------------------------------------------------------------

#define Bsz     2
#define Lseq    2048
#define Dmodel  768
#define Dinner  1536
#define Nstate  16
#define Rrank   48
#define Mrows   (Bsz * Lseq)        // 4096
#define XDBLW   (Rrank + 2*Nstate)  // 80

#define KB      8                   // K-rows per TDM-staged block
#define LROWS   72                  // padded LDS row stride (64 + 8 pad DWORDs)
                                    // 2*LROWS % 64 == 16 -> conflict-free halves

typedef float v2f __attribute__((ext_vector_type(2)));
typedef float v8f __attribute__((ext_vector_type(8)));
typedef unsigned int u32x4 __attribute__((ext_vector_type(4)));
typedef int i32x4 __attribute__((ext_vector_type(4)));
typedef int i32x8 __attribute__((ext_vector_type(8)));

#if defined(__has_builtin)
#if __has_builtin(__builtin_amdgcn_tensor_load_to_lds)
#define HAVE_TDM 1
#endif
#if __has_builtin(__builtin_amdgcn_s_wait_tensorcnt)
#define HAVE_TCNT 1
#endif
#endif

__device__ __forceinline__ v8f wmma_f32_k4(v2f a, v2f b, v8f c) {
    return __builtin_amdgcn_wmma_f32_16x16x4_f32(
        /*neg_a=*/false, a, /*neg_b=*/false, b,
        /*c_mod=*/(short)0, c, /*reuse_a=*/false, /*reuse_b=*/false);
}

__device__ __forceinline__ float siluf(float x) {
    // x * 1/(1+e^-x): v_exp_f32 + v_rcp_f32, no IEEE-div fixup sequence.
    return x * __builtin_amdgcn_rcpf(1.0f + __expf(-x));
}
__device__ __forceinline__ float softplusf(float x) {
    return (x > 20.0f) ? x : log1pf(__expf(x));
}

// --------------------------- TDM helpers -----------------------------------
__device__ __forceinline__ void wait_tensorcnt0() {
#if defined(HAVE_TCNT)
    __builtin_amdgcn_s_wait_tensorcnt((short)0);
#endif
    asm volatile("" ::: "memory");   // LDS now holds DMA'd data
}
__device__ __forceinline__ void wait_tensorcnt1() {
#if defined(HAVE_TCNT)
    __builtin_amdgcn_s_wait_tensorcnt((short)1);
#endif
    asm volatile("" ::: "memory");
}

// DMA one KBx64 f32 tile (row stride = strideElems) from global into LDS at
// byte offset ldsByte, padding each 64-DWORD row by 8 DWORDs (LROWS stride).
__device__ __forceinline__ void tdm_load_tile(const float* gsrc,
                                              unsigned ldsByte,
                                              unsigned strideElems) {
#if defined(HAVE_TDM)
    const unsigned long long ga = (unsigned long long)(size_t)gsrc;
    u32x4 g0;
    g0[0] = 1u;                                   // count=1 (valid user D#)
    g0[1] = ldsByte;                              // lds_addr
    g0[2] = (unsigned)(ga & 0xFFFFFFFFu);         // global_addr[31:0]
    g0[3] = (unsigned)((ga >> 32) & 0x01FFFFFFu)  // global_addr[56:32]
          | (2u << 30);                           // type=2 ("image")
    i32x8 g1;
    g1[0] = (2 << 16)                             // data_size = 4B
          | (1 << 20)                             // pad_enable
          | (5 << 22)                             // pad_interval: 64 DWORDs
          | (7 << 25);                            // pad_amount:   8 DWORDs
    g1[1] = (int)(64u << 16);                     // tensor_dim0[15:0] = 64
    g1[2] = (int)((unsigned)KB << 16);            // tensor_dim1[15:0] = KB
    g1[3] = (int)(64u << 16);                     // tile_dim0 = 64
    g1[4] = KB;                                   // tile_dim1 = KB, tile_dim2=0
    g1[5] = (int)strideElems;                     // tensor_dim0_stride lo32
    g1[6] = 0;                                    // stride0 hi16 | stride1 lo16
    g1[7] = 0;
    i32x4 z4 = {0, 0, 0, 0};
#if __clang_major__ >= 23
    i32x8 z8 = {0, 0, 0, 0, 0, 0, 0, 0};
    __builtin_amdgcn_tensor_load_to_lds(g0, g1, z4, z4, z8, 0);
#else
    __builtin_amdgcn_tensor_load_to_lds(g0, g1, z4, z4, 0);
#endif
#endif
}

__device__ __forceinline__ unsigned lds_byte_off(const void* p) {
    return (unsigned)(size_t)p;   // low 32 bits of generic VA = LDS offset
}

// ---------------------------------------------------------------------------
// Kernel 1: xz = x @ W_in ; xp = silu(xz[:, :1536]) ; zs = silu(xz[:, 1536:])
// M=4096, N=3072, K=768.  Wave -> 16x64 strip, TDM-staged B. grid(6,256).
// ---------------------------------------------------------------------------
__global__ __launch_bounds__(256) void k_gemm_in(
    const float* __restrict__ X, const float* __restrict__ W,
    float* __restrict__ xp, float* __restrict__ zs)
{
    const int lane  = threadIdx.x & 31;
    const int wave  = threadIdx.x >> 5;
    const int lrow  = lane & 15;
    const int lhalf = lane >> 4;
    const int row0  = blockIdx.y * 16;
    const int col0  = (blockIdx.x * 8 + wave) * 64;
    const int lda = Dmodel, ldb = 2 * Dinner;

    v8f acc[4] = {};
    const float* arow = X + (size_t)(row0 + lrow) * lda + 2 * lhalf;

#if defined(HAVE_TDM)
    __shared__ float ldsW[8 * 2 * KB * LROWS];
    float* buf0 = &ldsW[(wave * 2 + 0) * (KB * LROWS)];
    float* buf1 = &ldsW[(wave * 2 + 1) * (KB * LROWS)];
    const unsigned off0 = lds_byte_off(buf0);
    const unsigned off1 = lds_byte_off(buf1);

    const int NKB = Dmodel / KB;                       // 96
    tdm_load_tile(W + col0, off0, ldb);
    for (int kb = 0; kb < NKB; ++kb) {
        const float* cur = (kb & 1) ? buf1 : buf0;
        if (kb + 1 < NKB) {
            tdm_load_tile(W + (size_t)(kb + 1) * KB * ldb + col0,
                          (kb & 1) ? off0 : off1, ldb);
            wait_tensorcnt1();                         // current tile landed
        } else {
            wait_tensorcnt0();
        }
        #pragma unroll
        for (int kl = 0; kl < KB; kl += 4) {
            v2f a = *(const v2f*)(arow + kb * KB + kl);
            const float* bk = cur + (kl + 2 * lhalf) * LROWS + lrow;
            #pragma unroll
            for (int t = 0; t < 4; ++t) {
                v2f b; b.x = bk[t * 16]; b.y = bk[LROWS + t * 16];
                acc[t] = wmma_f32_k4(a, b, acc[t]);
            }
        }
    }
#else
    const float* bcol = W + (size_t)(2 * lhalf) * ldb + col0 + lrow;
    #pragma unroll 4
    for (int k = 0; k < Dmodel; k += 4) {
        v2f a = *(const v2f*)(arow + k);
        const float* bk = bcol + (size_t)k * ldb;
        #pragma unroll
        for (int t = 0; t < 4; ++t) {
            v2f b; b.x = bk[t * 16]; b.y = bk[(size_t)ldb + t * 16];
            acc[t] = wmma_f32_k4(a, b, acc[t]);
        }
    }
#endif

    #pragma unroll
    for (int t = 0; t < 4; ++t) {
        const int c = col0 + t * 16 + lrow;
        #pragma unroll
        for (int v = 0; v < 8; ++v) {
            const int r = row0 + v + 8 * lhalf;
            const float s = siluf(acc[t][v]);
            if (c < Dinner) xp[(size_t)r * Dinner + c] = s;
            else            zs[(size_t)r * Dinner + (c - Dinner)] = s;
        }
    }
}

// ---------------------------------------------------------------------------
// Kernel 2: x_dbl = xp @ W_x.  M=4096, N=80, K=1536.
// Wave -> one 16x16 tile. grid(5,32), block 256.
// ---------------------------------------------------------------------------
__global__ __launch_bounds__(256) void k_gemm_xdbl(
    const float* __restrict__ XP, const float* __restrict__ Wx,
    float* __restrict__ xdbl)
{
    const int lane  = threadIdx.x & 31;
    const int wave  = threadIdx.x >> 5;
    const int lrow  = lane & 15;
    const int lhalf = lane >> 4;
    const int row0  = (blockIdx.y * 8 + wave) * 16;
    const int col0  = blockIdx.x * 16;
    const int lda = Dinner, ldb = XDBLW;

    v8f acc = {};
    const float* arow = XP + (size_t)(row0 + lrow) * lda + 2 * lhalf;
    const float* bcol = Wx + (size_t)(2 * lhalf) * ldb + col0 + lrow;

    #pragma unroll 8
    for (int k = 0; k < Dinner; k += 4) {
        v2f a = *(const v2f*)(arow + k);
        v2f b;
        b.x = bcol[(size_t)k * ldb];
        b.y = bcol[(size_t)(k + 1) * ldb];
        acc = wmma_f32_k4(a, b, acc);
    }

    #pragma unroll
    for (int v = 0; v < 8; ++v)
        xdbl[(size_t)(row0 + v + 8 * lhalf) * XDBLW + col0 + lrow] = acc[v];
}

// ---------------------------------------------------------------------------
// Kernel 3: dt = softplus(x_dbl[:, :48] @ W_dt + b_dt).  M=4096, N=1536, K=48.
// Wave -> 16x64 strip. grid(3,256), block 256.
// ---------------------------------------------------------------------------
__global__ __launch_bounds__(256) void k_gemm_dt(
    const float* __restrict__ XD, const float* __restrict__ Wdt,
    const float* __restrict__ bdt, float* __restrict__ dt)
{
    const int lane  = threadIdx.x & 31;
    const int wave  = threadIdx.x >> 5;
    const int lrow  = lane & 15;
    const int lhalf = lane >> 4;
    const int row0  = blockIdx.y * 16;
    const int col0  = (blockIdx.x * 8 + wave) * 64;
    const int lda = XDBLW, ldb = Dinner;

    v8f acc[4] = {};
    const float* arow = XD + (size_t)(row0 + lrow) * lda + 2 * lhalf;
    const float* bcol = Wdt + (size_t)(2 * lhalf) * ldb + col0 + lrow;

    #pragma unroll
    for (int k = 0; k < Rrank; k += 4) {
        v2f a = *(const v2f*)(arow + k);
        const float* bk = bcol + (size_t)k * ldb;
        #pragma unroll
        for (int t = 0; t < 4; ++t) {
            v2f b; b.x = bk[t * 16]; b.y = bk[(size_t)ldb + t * 16];
            acc[t] = wmma_f32_k4(a, b, acc[t]);
        }
    }

    #pragma unroll
    for (int t = 0; t < 4; ++t) {
        const int c = col0 + t * 16 + lrow;
        const float bias = bdt[c];
        #pragma unroll
        for (int v = 0; v < 8; ++v) {
            const int r = row0 + v + 8 * lhalf;
            dt[(size_t)r * Dinner + c] = softplusf(acc[t][v] + bias);
        }
    }
}

// ---------------------------------------------------------------------------
// Kernel 4: selective scan. Thread <-> (b, d) channel; h[16] in registers.
// ymid = (scan_y + D_skip*xp) * zs.  grid(24), block 128.
// ---------------------------------------------------------------------------
__global__ __launch_bounds__(128) void k_scan(
    const float* __restrict__ dt, const float* __restrict__ xdbl,
    const float* __restrict__ xp, const float* __restrict__ zs,
    const float* __restrict__ A_log, const float* __restrict__ D_skip,
    float* __restrict__ ymid)
{
    const int blocksPerB = Dinner / 128;            // 12
    const int b = blockIdx.x / blocksPerB;
    const int d = (blockIdx.x % blocksPerB) * 128 + threadIdx.x;

    float Ad[Nstate];
    #pragma unroll
    for (int n = 0; n < Nstate; ++n)
        Ad[n] = -__expf(A_log[(size_t)d * Nstate + n]);
    const float Dv = D_skip[d];

    float h[Nstate];
    #pragma unroll
    for (int n = 0; n < Nstate; ++n) h[n] = 0.0f;

    const size_t rowBase = (size_t)b * Lseq;
    for (int t = 0; t < Lseq; ++t) {
        const size_t r = rowBase + t;
        const float dtv = dt[r * Dinner + d];
        const float xv  = xp[r * Dinner + d];
        const float zv  = zs[r * Dinner + d];

        // B (16) and C (16) uniform across the wave: broadcast float4 loads.
        const float4* bc = (const float4*)(xdbl + r * XDBLW + Rrank);
        float BC[32];
        #pragma unroll
        for (int i = 0; i < 8; ++i) {
            const float4 v = bc[i];
            BC[4*i+0] = v.x; BC[4*i+1] = v.y; BC[4*i+2] = v.z; BC[4*i+3] = v.w;
        }

        const float dx = dtv * xv;
        float y = 0.0f;
        #pragma unroll
        for (int n = 0; n < Nstate; ++n) {
            const float dA = __expf(dtv * Ad[n]);
            h[n] = dA * h[n] + dx * BC[n];
            y += h[n] * BC[Nstate + n];
        }
        ymid[r * Dinner + d] = (y + Dv * xv) * zv;
    }
}

// ---------------------------------------------------------------------------
// Kernel 5: hres = x + ymid @ W_out.  M=4096, N=768, K=1536.
// 4 waves/block, wave -> 16x64 strip (distinct N-strip => distinct TDM tile).
// grid(3,256), block 128.
// ---------------------------------------------------------------------------
__global__ __launch_bounds__(128) void k_gemm_out(
    const float* __restrict__ Y, const float* __restrict__ Wout,
    const float* __restrict__ X, float* __restrict__ hres)
{
    const int lane  = threadIdx.x & 31;
    const int wave  = threadIdx.x >> 5;
    const int lrow  = lane & 15;
    const int lhalf = lane >> 4;
    const int row0  = blockIdx.y * 16;
    const int col0  = (blockIdx.x * 4 + wave) * 64;
    const int lda = Dinner, ldb = Dmodel;

    v8f acc[4] = {};
    const float* arow = Y + (size_t)(row0 + lrow) * lda + 2 * lhalf;

#if defined(HAVE_TDM)
    __shared__ float ldsW[4 * 2 * KB * LROWS];
    float* buf0 = &ldsW[(wave * 2 + 0) * (KB * LROWS)];
    float* buf1 = &ldsW[(wave * 2 + 1) * (KB * LROWS)];
    const unsigned off0 = lds_byte_off(buf0);
    const unsigned off1 = lds_byte_off(buf1);

    const int NKB = Dinner / KB;                       // 192
    tdm_load_tile(Wout + col0, off0, ldb);
    for (int kb = 0; kb < NKB; ++kb) {
        const float* cur = (kb & 1) ? buf1 : buf0;
        if (kb + 1 < NKB) {
            tdm_load_tile(Wout + (size_t)(kb + 1) * KB * ldb + col0,
                          (kb & 1) ? off0 : off1, ldb);
            wait_tensorcnt1();
        } else {
            wait_tensorcnt0();
        }
        #pragma unroll
        for (int kl = 0; kl < KB; kl += 4) {
            v2f a = *(const v2f*)(arow + kb * KB + kl);
            const float* bk = cur + (kl + 2 * lhalf) * LROWS + lrow;
            #pragma unroll
            for (int t = 0; t < 4; ++t) {
                v2f b; b.x = bk[t * 16]; b.y = bk[LROWS + t * 16];
                acc[t] = wmma_f32_k4(a, b, acc[t]);
            }
        }
    }
#else
    const float* bcol = Wout + (size_t)(2 * lhalf) * ldb + col0 + lrow;
    #pragma unroll 4
    for (int k = 0; k < Dinner; k += 4) {
        v2f a = *(const v2f*)(arow + k);
        const float* bk = bcol + (size_t)k * ldb;
        #pragma unroll
        for (int t = 0; t < 4; ++t) {
            v2f b; b.x = bk[t * 16]; b.y = bk[(size_t)ldb + t * 16];
            acc[t] = wmma_f32_k4(a, b, acc[t]);
        }
    }
#endif

    #pragma unroll
    for (int t = 0; t < 4; ++t) {
        const int c = col0 + t * 16 + lrow;
        #pragma unroll
        for (int v = 0; v < 8; ++v) {
            const int r = row0 + v + 8 * lhalf;
            hres[(size_t)r * Dmodel + c] = X[(size_t)r * Dmodel + c] + acc[t][v];
        }
    }
}

// ---------------------------------------------------------------------------
// Kernel 6: LayerNorm over D=768 per row. grid(4096), block 256 (3 elems/thr).
// ---------------------------------------------------------------------------
__global__ __launch_bounds__(256) void k_ln(
    const float* __restrict__ hres, const float* __restrict__ gamma,
    const float* __restrict__ beta, float* __restrict__ out)
{
    const int row = blockIdx.x;
    const int tid = threadIdx.x;
    const float* hr = hres + (size_t)row * Dmodel;

    const float v0 = hr[tid], v1 = hr[tid + 256], v2 = hr[tid + 512];
    float s  = v0 + v1 + v2;
    float ss = v0*v0 + v1*v1 + v2*v2;

    __shared__ float sSum[256];
    __shared__ float sSq[256];
    sSum[tid] = s; sSq[tid] = ss;
    __syncthreads();
    #pragma unroll
    for (int off = 128; off > 0; off >>= 1) {
        if (tid < off) { sSum[tid] += sSum[tid + off]; sSq[tid] += sSq[tid + off]; }
        __syncthreads();
    }
    const float mu  = sSum[0] * (1.0f / Dmodel);
    const float var = sSq[0] * (1.0f / Dmodel) - mu * mu;
    const float inv = rsqrtf(var + 1e-5f);

    float* orow = out + (size_t)row * Dmodel;
    orow[tid      ] = (v0 - mu) * inv * gamma[tid      ] + beta[tid      ];
    orow[tid + 256] = (v1 - mu) * inv * gamma[tid + 256] + beta[tid + 256];
    orow[tid + 512] = (v2 - mu) * inv * gamma[tid + 512] + beta[tid + 512];
}

// ---------------------------------------------------------------------------
extern "C" void kernel_launch(void* const* d_in, const int* in_sizes, int n_in,
                              void* d_out, int out_size, void* d_ws, size_t ws_size,
                              hipStream_t stream) {
    const float* x      = (const float*)d_in[0];
    const float* W_in   = (const float*)d_in[1];
    const float* W_x    = (const float*)d_in[2];
    const float* W_dt   = (const float*)d_in[3];
    const float* b_dt   = (const float*)d_in[4];
    const float* A_log  = (const float*)d_in[5];
    const float* D_skip = (const float*)d_in[6];
    const float* W_out  = (const float*)d_in[7];
    const float* gamma  = (const float*)d_in[8];
    const float* beta   = (const float*)d_in[9];
    float* out = (float*)d_out;

    // Workspace layout (floats)
    float* ws   = (float*)d_ws;
    float* xp   = ws;                                  // 4096*1536
    float* zs   = xp   + (size_t)Mrows * Dinner;       // 4096*1536
    float* xdbl = zs   + (size_t)Mrows * Dinner;       // 4096*80
    float* dt   = xdbl + (size_t)Mrows * XDBLW;        // 4096*1536
    float* ymid = dt   + (size_t)Mrows * Dinner;       // 4096*1536
    float* hres = ymid + (size_t)Mrows * Dinner;       // 4096*768

    k_gemm_in  <<<dim3(6, Mrows / 16), 256, 0, stream>>>(x, W_in, xp, zs);
    k_gemm_xdbl<<<dim3(5, Mrows / 128), 256, 0, stream>>>(xp, W_x, xdbl);
    k_gemm_dt  <<<dim3(3, Mrows / 16), 256, 0, stream>>>(xdbl, W_dt, b_dt, dt);
    k_scan     <<<dim3(Bsz * (Dinner / 128)), 128, 0, stream>>>(dt, xdbl, xp, zs,
                                                                A_log, D_skip, ymid);
    k_gemm_out <<<dim3(3, Mrows / 16), 128, 0, stream>>>(ymid, W_out, x, hres);
    k_ln       <<<dim3(Mrows), 256, 0, stream>>>(hres, gamma, beta, out);
}